// DiffeomorphicTransform_70660801954600
// MI455X (gfx1250) — compile-verified
//
#include <hip/hip_runtime.h>

// Diffeomorphic scaling-and-squaring, 192^3 x 3 float32 flow, 7 steps.
// Gather/bandwidth-bound: no matrix math, so no WMMA. Strategy:
//  - repack [3,D,H,W] -> interleaved [D,H,W,3] so each trilinear corner is one
//    contiguous 12B gather (global_load_b96) instead of 3 scattered b32s
//  - ping-pong two 85MB interleaved buffers (d_ws + d_out-as-scratch); the
//    pair (170MB) fits in MI455X's 192MB L2, so the increasingly incoherent
//    gathers of later squaring steps stay on-chip
//  - temporal hints: NT loads for read-once streams (pack input, unpack
//    last-use input), NT stores for the write-once final output, default RT
//    for the L2-resident ping-pong traffic; global_prefetch_b8 on the
//    streaming self-read in compose

#define DZ 192
#define HY 192
#define WX 192
#define NVOX (DZ * HY * WX)
#define HALF_EXTENT 95.5f   // (W-1)/2, same for all axes
#define FMAX_COORD 191.0f

__device__ __forceinline__ float lerpf(float a, float b, float t) {
    return fmaf(t, b - a, a);
}

__device__ __forceinline__ void ld3(const float* __restrict__ p,
                                    float& a, float& b, float& c) {
    a = p[0]; b = p[1]; c = p[2];
}

// ---- channel-major [3,N] -> interleaved [N,3], scaled by 1/2^7 -------------
__global__ __launch_bounds__(WX) void pack_kernel(const float* __restrict__ in,
                                                  float* __restrict__ out) {
    const int x = threadIdx.x;
    const int y = blockIdx.x;
    const int z = blockIdx.y;
    const int idx = (z * HY + y) * WX + x;
    const float s = 1.0f / 128.0f;
    // read-once HBM stream: NT so it doesn't evict the L2 working set
    const float d = __builtin_nontemporal_load(in + idx) * s;
    const float h = __builtin_nontemporal_load(in + NVOX + idx) * s;
    const float w = __builtin_nontemporal_load(in + 2 * NVOX + idx) * s;
    float* __restrict__ p = out + idx * 3;
    p[0] = d; p[1] = h; p[2] = w;
}

// ---- one squaring step: nxt = cur + trilinear(cur, id + cur) ---------------
__global__ __launch_bounds__(WX) void compose_kernel(const float* __restrict__ cur,
                                                     float* __restrict__ nxt) {
    const int x = threadIdx.x;
    const int y = blockIdx.x;
    const int z = blockIdx.y;
    const int idx = (z * HY + y) * WX + x;

    const float* __restrict__ op = cur + idx * 3;
    // prefetch the streaming self-read a few rows ahead (speculative; safe)
    __builtin_prefetch(cur + (idx + 8 * WX) * 3, 0, 1);

    float od, oh, ow;
    ld3(op, od, oh, ow);

    // sample position in voxel units: pos + disp*(size-1)/2, border-clamped
    const float fx = fminf(fmaxf(fmaf(ow, HALF_EXTENT, (float)x), 0.0f), FMAX_COORD);
    const float fy = fminf(fmaxf(fmaf(oh, HALF_EXTENT, (float)y), 0.0f), FMAX_COORD);
    const float fz = fminf(fmaxf(fmaf(od, HALF_EXTENT, (float)z), 0.0f), FMAX_COORD);

    const float x0f = floorf(fx), y0f = floorf(fy), z0f = floorf(fz);
    const float wx = fx - x0f, wy = fy - y0f, wz = fz - z0f;

    const int x0 = (int)x0f, y0 = (int)y0f, z0 = (int)z0f;
    const int x1 = min(x0 + 1, WX - 1);
    const int y1 = min(y0 + 1, HY - 1);
    const int z1 = min(z0 + 1, DZ - 1);

    const int r00 = (z0 * HY + y0) * WX;
    const int r01 = (z0 * HY + y1) * WX;
    const int r10 = (z1 * HY + y0) * WX;
    const int r11 = (z1 * HY + y1) * WX;

    // 8 corner gathers, each a contiguous 12B (b96) access; default RT so the
    // L2-resident field stays hot for subsequent gathers
    float d000,h000,w000, d001,h001,w001, d010,h010,w010, d011,h011,w011;
    float d100,h100,w100, d101,h101,w101, d110,h110,w110, d111,h111,w111;
    ld3(cur + (r00 + x0) * 3, d000, h000, w000);
    ld3(cur + (r00 + x1) * 3, d001, h001, w001);
    ld3(cur + (r01 + x0) * 3, d010, h010, w010);
    ld3(cur + (r01 + x1) * 3, d011, h011, w011);
    ld3(cur + (r10 + x0) * 3, d100, h100, w100);
    ld3(cur + (r10 + x1) * 3, d101, h101, w101);
    ld3(cur + (r11 + x0) * 3, d110, h110, w110);
    ld3(cur + (r11 + x1) * 3, d111, h111, w111);

    // trilinear: 7 fma-lerps per channel
    float sd, sh, sw;
    {
        const float a00 = lerpf(d000, d001, wx), a01 = lerpf(d010, d011, wx);
        const float a10 = lerpf(d100, d101, wx), a11 = lerpf(d110, d111, wx);
        sd = lerpf(lerpf(a00, a01, wy), lerpf(a10, a11, wy), wz);
    }
    {
        const float a00 = lerpf(h000, h001, wx), a01 = lerpf(h010, h011, wx);
        const float a10 = lerpf(h100, h101, wx), a11 = lerpf(h110, h111, wx);
        sh = lerpf(lerpf(a00, a01, wy), lerpf(a10, a11, wy), wz);
    }
    {
        const float a00 = lerpf(w000, w001, wx), a01 = lerpf(w010, w011, wx);
        const float a10 = lerpf(w100, w101, wx), a11 = lerpf(w110, w111, wx);
        sw = lerpf(lerpf(a00, a01, wy), lerpf(a10, a11, wy), wz);
    }

    // RT store: this buffer is the next iteration's gather target (keep in L2)
    float* __restrict__ q = nxt + idx * 3;
    q[0] = od + sd;
    q[1] = oh + sh;
    q[2] = ow + sw;
}

// ---- interleaved [N,3] -> channel-major [3,N], NT both directions ----------
__global__ __launch_bounds__(WX) void unpack_kernel(const float* __restrict__ in,
                                                    float* __restrict__ out) {
    const int x = threadIdx.x;
    const int y = blockIdx.x;
    const int z = blockIdx.y;
    const int idx = (z * HY + y) * WX + x;
    const float* __restrict__ p = in + idx * 3;
    // last-use read of the final field: NT so it doesn't get re-promoted
    const float d = __builtin_nontemporal_load(p + 0);
    const float h = __builtin_nontemporal_load(p + 1);
    const float w = __builtin_nontemporal_load(p + 2);
    // write-once output: NT stores
    __builtin_nontemporal_store(d, out + idx);
    __builtin_nontemporal_store(h, out + NVOX + idx);
    __builtin_nontemporal_store(w, out + 2 * NVOX + idx);
}

extern "C" void kernel_launch(void* const* d_in, const int* in_sizes, int n_in,
                              void* d_out, int out_size, void* d_ws, size_t ws_size,
                              hipStream_t stream) {
    (void)in_sizes; (void)n_in; (void)out_size; (void)ws_size;

    const float* flow = (const float*)d_in[0];
    float* out = (float*)d_out;

    // Two interleaved ping-pong buffers, each 3*NVOX floats (~81 MiB):
    //   bufA lives in d_ws; bufB reuses d_out's memory as raw scratch
    //   (out_size == 3*NVOX, identical footprint). Together they fit in the
    //   192 MB L2, so squaring-step gathers stay on-chip.
    float* bufA = (float*)d_ws;
    float* bufB = out;

    const dim3 grid(HY, DZ);
    const dim3 block(WX);

    pack_kernel<<<grid, block, 0, stream>>>(flow, bufB);

    float* cur = bufB;
    float* nxt = bufA;
    for (int t = 0; t < 7; ++t) {
        compose_kernel<<<grid, block, 0, stream>>>(cur, nxt);
        float* tmp = cur; cur = nxt; nxt = tmp;
    }
    // 7 (odd) steps starting from bufB -> final field sits in bufA (d_ws),
    // so unpack can safely write channel-major output over d_out.
    unpack_kernel<<<grid, block, 0, stream>>>(cur, out);
}